// DyanEncoder_62448824484580
// MI455X (gfx1250) — compile-verified
//
#include <hip/hip_runtime.h>
#include <hip/hip_bf16.h>
#include <math.h>

typedef __attribute__((ext_vector_type(2))) float v2f;
typedef __attribute__((ext_vector_type(8))) float v8f;

#define T_SEQ  36
#define NPOLE  80
#define P_DIC  161
#define DD     50
#define NB     64
#define LAM    0.1f
#define MAXIT  100

#define M_PAD  192      // 12 row tiles of 16  -> 48 tiles = 8 waves * 6, uniform
#define K_PAD  164      // 41 K-steps of 4
#define N_PAD  64       // 4 col tiles of 16
#define NWAVES 8
#define TPW    6        // m-tiles per wave, exact for every wave

// workspace layout (float offsets)
#define WS_LINV 0
#define WS_A    16
#define WS_DN   (16 + M_PAD * K_PAD)
#define WS_D    (WS_DN + T_SEQ * P_DIC)

// -------- Kernel 1: build D, Dn, A = I - Dn^T Dn * Linv --------
__global__ void __launch_bounds__(256) dyan_setup(const float* __restrict__ rr,
                                                  const float* __restrict__ theta,
                                                  float* __restrict__ ws,
                                                  float* __restrict__ outD) {
  __shared__ float sD[T_SEQ * P_DIC];
  __shared__ float snorm[P_DIC];
  __shared__ float sfro;
  const int tid = threadIdx.x;

  if (tid == 0) sfro = 0.f;
  // build dictionary D: row t = [1, r^t cos(t*th), r^t sin(t*th)]
  for (int idx = tid; idx < T_SEQ * P_DIC; idx += 256) {
    int t = idx / P_DIC, p = idx % P_DIC;
    float v;
    if (p == 0) {
      v = 1.f;
    } else if (p <= NPOLE) {
      int j = p - 1;
      v = powf(rr[j], (float)t) * cosf((float)t * theta[j]);
    } else {
      int j = p - 1 - NPOLE;
      v = powf(rr[j], (float)t) * sinf((float)t * theta[j]);
    }
    sD[idx] = v;
    ws[WS_D + idx] = v;
    outD[idx] = v;        // D output (written once, by this single block)
  }
  __syncthreads();
  // column norms
  for (int p = tid; p < P_DIC; p += 256) {
    float s = 0.f;
    for (int t = 0; t < T_SEQ; ++t) { float d = sD[t * P_DIC + p]; s += d * d; }
    snorm[p] = sqrtf(s);
  }
  __syncthreads();
  // normalize -> Dn
  for (int idx = tid; idx < T_SEQ * P_DIC; idx += 256) {
    float v = sD[idx] / snorm[idx % P_DIC];
    sD[idx] = v;
    ws[WS_DN + idx] = v;
  }
  __syncthreads();
  // DtD (temp in A region) + Frobenius norm
  float fro = 0.f;
  for (int e = tid; e < P_DIC * P_DIC; e += 256) {
    int p = e / P_DIC, q = e % P_DIC;
    float s = 0.f;
    for (int t = 0; t < T_SEQ; ++t) s += sD[t * P_DIC + p] * sD[t * P_DIC + q];
    ws[WS_A + p * K_PAD + q] = s;
    fro += s * s;
  }
  atomicAdd(&sfro, fro);
  __syncthreads();
  const float Linv = 1.0f / sqrtf(sfro);
  if (tid == 0) ws[WS_LINV] = Linv;
  __syncthreads();
  // A = I - DtD * Linv, zero padding (rows/cols 161..191/163)
  for (int e = tid; e < M_PAD * K_PAD; e += 256) {
    int p = e / K_PAD, q = e % K_PAD;
    float v = 0.f;
    if (p < P_DIC && q < P_DIC)
      v = ((p == q) ? 1.f : 0.f) - ws[WS_A + e] * Linv;
    ws[WS_A + e] = v;
  }
}

// -------- Kernel 2: per-batch reweighted FISTA with f32 WMMA --------
// Wave w owns column-tile nt = w&3 and m-tiles mt = (w>>2) + 2*ti, ti=0..5.
// 12x4 tile grid = 48 tiles = 8 waves * 6: every wave has exactly 6 tiles,
// so the hot loop is branch-free (WMMA runs with full EXEC, no phi copies).
// k-outer / m-inner: B operand loaded once per k-step, reused across all
// m-tiles. The k-loop is fully unrolled: no loop-carried accumulator phis,
// all LDS addresses become immediate offset0/offset1 forms, and the
// scheduler can pipeline ds loads across the whole 246-WMMA body.
__global__ void __launch_bounds__(256) dyan_fista(const float* __restrict__ xg,
                                                  const float* __restrict__ ws,
                                                  float* __restrict__ outCode,
                                                  float* __restrict__ outRec) {
  extern __shared__ float smem[];
  float* A_lds = smem;                          // M_PAD*K_PAD
  float* yb0   = smem + M_PAD * K_PAD;          // M_PAD*N_PAD
  float* yb1   = yb0 + M_PAD * N_PAD;           // M_PAD*N_PAD
  float* S     = yb1 + M_PAD * N_PAD;           // N_PAD column sums

  const int n    = blockIdx.x;
  const int tid  = threadIdx.x;
  const int wid  = tid >> 5;
  const int lane = tid & 31;
  const int half = lane >> 4;     // 0 or 1
  const int col  = lane & 15;     // N index within tile
  const int mrow = lane & 15;     // A-operand row within tile
  const int mt0  = wid >> 2;      // 0 (even m-tiles) or 1 (odd m-tiles)
  const int nt   = wid & 3;       // fixed column tile per wave

  const float  Linv = ws[WS_LINV];
  const float* Ag   = ws + WS_A;
  const float* Dn   = ws + WS_DN;
  const float* Dg   = ws + WS_D;
  const float* xb   = xg + (size_t)n * (T_SEQ * DD);

  for (int i = tid; i < M_PAD * K_PAD; i += 256) A_lds[i] = Ag[i];
  for (int i = tid; i < M_PAD * N_PAD; i += 256) { yb0[i] = 0.f; yb1[i] = 0.f; }

  // Per-wave FISTA state held in WMMA C-layout registers.
  v8f xold[TPW], dty[TPW], wl[TPW];
  #pragma unroll
  for (int ti = 0; ti < TPW; ++ti) {
    const int mt = mt0 + 2 * ti;
    #pragma unroll
    for (int r = 0; r < 8; ++r) {
      float vd = 0.f, vw = 0.f;
      const int p = mt * 16 + r + 8 * half;
      const int d = nt * 16 + col;
      if (p < P_DIC && d < DD) {
        float s = 0.f;
        for (int t = 0; t < T_SEQ; ++t) s += Dn[t * P_DIC + p] * xb[t * DD + d];
        vd = s * Linv;           // DtY * Linv
        vw = LAM * Linv;         // w = 1 initially
      }
      dty[ti][r] = vd;
      wl[ti][r]  = vw;
      xold[ti][r] = 0.f;
    }
  }
  __syncthreads();

  float* ycur = yb0;
  float* ynxt = yb1;
  const v8f vzero = {};

  for (int round = 0; round < 2; ++round) {
    float tcur = 1.f;
    #pragma unroll 1
    for (int it = 0; it < MAXIT; ++it) {
      const float tnext = 0.5f * (1.f + sqrtf(1.f + 4.f * tcur * tcur));
      const float tt = (tcur - 1.f) / tnext;
      tcur = tnext;

      v8f acc[TPW];
      #pragma unroll
      for (int ti = 0; ti < TPW; ++ti) acc[ti] = vzero;

      // A-operand: lane&15 = M, VGPR0/1 = K{0,1} (lanes 0-15) / K{2,3} (16-31)
      const float* Ab = A_lds + (mt0 * 16 + mrow) * K_PAD + 2 * half;
      // B-operand: lane&15 = N, VGPR0/1 = K{0,1} (lanes 0-15) / K{2,3} (16-31)
      const float* Bc = ycur + nt * 16 + col + half * (2 * N_PAD);

      #pragma unroll
      for (int k0 = 0; k0 < K_PAD; k0 += 4) {
        v2f b;
        b.x = Bc[k0 * N_PAD];
        b.y = Bc[k0 * N_PAD + N_PAD];
        #pragma unroll
        for (int ti = 0; ti < TPW; ++ti) {
          v2f a;
          a.x = Ab[ti * 32 * K_PAD + k0];       // mt = mt0 + 2*ti
          a.y = Ab[ti * 32 * K_PAD + k0 + 1];
          acc[ti] = __builtin_amdgcn_wmma_f32_16x16x4_f32(
              false, a, false, b, (short)0, acc[ti], false, false);
        }
      }

      // fused soft-threshold + momentum in C layout
      #pragma unroll
      for (int ti = 0; ti < TPW; ++ti) {
        const int mt = mt0 + 2 * ti;
        #pragma unroll
        for (int r = 0; r < 8; ++r) {
          const float z  = acc[ti][r] + dty[ti][r];
          const float w  = wl[ti][r];
          const float xn = fmaxf(z - w, 0.f) + fminf(z + w, 0.f);
          const float yn = xn + tt * (xn - xold[ti][r]);
          xold[ti][r] = xn;
          const int p = mt * 16 + r + 8 * half;
          const int d = nt * 16 + col;
          ynxt[p * N_PAD + d] = yn;
        }
      }
      __syncthreads();
      float* tmp = ycur; ycur = ynxt; ynxt = tmp;
    }

    if (round == 0) {
      // reweight: w = wi / sum_p(wi) * P, wi = 1/(|code|+1e-6)
      for (int i = tid; i < N_PAD; i += 256) S[i] = 0.f;
      __syncthreads();
      #pragma unroll
      for (int ti = 0; ti < TPW; ++ti) {
        const int mt = mt0 + 2 * ti;
        #pragma unroll
        for (int r = 0; r < 8; ++r) {
          const int p = mt * 16 + r + 8 * half;
          const int d = nt * 16 + col;
          if (p < P_DIC && d < DD)
            atomicAdd(&S[d], 1.f / (fabsf(xold[ti][r]) + 1e-6f));
        }
      }
      __syncthreads();
      #pragma unroll
      for (int ti = 0; ti < TPW; ++ti) {
        const int mt = mt0 + 2 * ti;
        #pragma unroll
        for (int r = 0; r < 8; ++r) {
          float vw = 0.f;
          const int p = mt * 16 + r + 8 * half;
          const int d = nt * 16 + col;
          if (p < P_DIC && d < DD) {
            const float wi = 1.f / (fabsf(xold[ti][r]) + 1e-6f);
            vw = (LAM * Linv) * ((float)P_DIC) * wi / S[d];
          }
          wl[ti][r]  = vw;
          xold[ti][r] = 0.f;     // FISTA restarts from x0 = 0
        }
      }
      __syncthreads();
      for (int i = tid; i < M_PAD * N_PAD; i += 256) ycur[i] = 0.f;
      __syncthreads();
    }
  }

  // write code + stage into LDS (padded layout) for reconst
  float* codeOut = outCode + (size_t)n * (P_DIC * DD);
  #pragma unroll
  for (int ti = 0; ti < TPW; ++ti) {
    const int mt = mt0 + 2 * ti;
    #pragma unroll
    for (int r = 0; r < 8; ++r) {
      const int p = mt * 16 + r + 8 * half;
      const int d = nt * 16 + col;
      const float v = xold[ti][r];
      if (p < P_DIC && d < DD) codeOut[p * DD + d] = v;
      ycur[p * N_PAD + d] = v;
    }
  }
  __syncthreads();
  // reconst = D @ code  (unnormalized D, per reference)
  float* recOut = outRec + (size_t)n * (T_SEQ * DD);
  for (int o = tid; o < T_SEQ * DD; o += 256) {
    const int t = o / DD, d = o % DD;
    float s = 0.f;
    for (int p = 0; p < P_DIC; ++p) s += Dg[t * P_DIC + p] * ycur[p * N_PAD + d];
    recOut[o] = s;
  }
}

extern "C" void kernel_launch(void* const* d_in, const int* in_sizes, int n_in,
                              void* d_out, int out_size, void* d_ws, size_t ws_size,
                              hipStream_t stream) {
  const float* x     = (const float*)d_in[0];   // [64,36,50]
  const float* rr    = (const float*)d_in[1];   // [80]
  const float* theta = (const float*)d_in[2];   // [80]
  float* ws  = (float*)d_ws;
  float* out = (float*)d_out;

  float* outCode = out;                              // 64*161*50
  float* outD    = out + (size_t)NB * P_DIC * DD;    // 36*161
  float* outRec  = outD + (size_t)T_SEQ * P_DIC;     // 64*36*50

  dyan_setup<<<1, 256, 0, stream>>>(rr, theta, ws, outD);

  const size_t shmem =
      (size_t)(M_PAD * K_PAD + 2 * M_PAD * N_PAD + N_PAD) * sizeof(float);
  dyan_fista<<<NB, 256, shmem, stream>>>(x, ws, outCode, outRec);
}